// SCBILinear_63367947485206
// MI455X (gfx1250) — compile-verified
//
#include <hip/hip_runtime.h>

typedef float v2f __attribute__((ext_vector_type(2)));
typedef float v8f __attribute__((ext_vector_type(8)));

#define FOLD 10000
#define D 512
#define O 256
#define NF 5
#define NL 5
#define NTR 40000.0f
#define NALL 50000.0f

__constant__ float c_lam[5] = {0.01f, 0.1f, 1.0f, 10.0f, 100.0f};

static __device__ inline v8f wmma4(v2f a, v2f b, v8f c) {
    // D = A(16x4) * B(4x16) + C, fp32 matrix pipe
    return __builtin_amdgcn_wmma_f32_16x16x4_f32(false, a, false, b, (short)0, c, false, false);
}

// ---------------------------------------------------------------------------
// Kernel 1: per-fold sufficient statistics  C_k = Xr[k]^T * [Xr[k] | Yr[k]]
// grid (6, 4, 5) = (n-tiles over 768, m-tiles over 512, fold); block 256 (8 waves)
// LDS panels are stored K-pair interleaved: As2[k/2][col][2] so a WMMA fragment
// {A[k][col], A[k+1][col]} is one aligned ds_load_b64 into an even VGPR pair.
// Global->LDS staging is software-pipelined through registers.
// ---------------------------------------------------------------------------
#define TS 128
#define KC 32
__global__ void stats_kernel(const float* __restrict__ X, const float* __restrict__ Y,
                             float* __restrict__ xxT, float* __restrict__ xyT) {
    const int fold = blockIdx.z;
    const int m0 = blockIdx.y * TS;
    const int n0g = blockIdx.x * TS;           // 0..767
    const bool isY = (n0g >= D);
    const float* Xf = X + (size_t)fold * FOLD * D;
    const float* Bsrc = isY ? (Y + (size_t)fold * FOLD * O) : Xf;
    const int ldb = isY ? O : D;
    const int bcol0 = isY ? (n0g - D) : n0g;

    // [K-pair][col][2], padded col dim rotates banks by 4 per K-pair row
    __shared__ float As2[KC / 2][TS + 2][2];
    __shared__ float Bs2[KC / 2][TS + 2][2];

    const int t = threadIdx.x;
    const int wave = t >> 5;
    const int lane = t & 31;
    const int half = lane >> 4;
    const int lm = lane & 15;
    const int wm = wave & 1;    // 2 groups of 64 rows
    const int wn = wave >> 1;   // 4 groups of 32 cols

    const int s0 = t, s1 = t + 256;            // two 256-slot passes
    const int rp0 = s0 >> 5, c40 = (s0 & 31) << 2;
    const int rp1 = s1 >> 5, c41 = (s1 & 31) << 2;

    v8f acc[4][2] = {};
    float4 ra[2][2], rb[2][2];                 // staging regs: [pass][row-in-pair]

    // prologue: fetch first stage
    ra[0][0] = *(const float4*)(Xf + (size_t)(2 * rp0) * D + m0 + c40);
    ra[0][1] = *(const float4*)(Xf + (size_t)(2 * rp0 + 1) * D + m0 + c40);
    ra[1][0] = *(const float4*)(Xf + (size_t)(2 * rp1) * D + m0 + c41);
    ra[1][1] = *(const float4*)(Xf + (size_t)(2 * rp1 + 1) * D + m0 + c41);
    rb[0][0] = *(const float4*)(Bsrc + (size_t)(2 * rp0) * ldb + bcol0 + c40);
    rb[0][1] = *(const float4*)(Bsrc + (size_t)(2 * rp0 + 1) * ldb + bcol0 + c40);
    rb[1][0] = *(const float4*)(Bsrc + (size_t)(2 * rp1) * ldb + bcol0 + c41);
    rb[1][1] = *(const float4*)(Bsrc + (size_t)(2 * rp1 + 1) * ldb + bcol0 + c41);

    for (int kk = 0; kk < FOLD; kk += KC) {
        __syncthreads();
        // interleaved store: pairs {row 2rp, row 2rp+1} -> ds_store_b64
        {
            const float* a0 = (const float*)&ra[0][0];
            const float* a1 = (const float*)&ra[0][1];
            const float* b0 = (const float*)&rb[0][0];
            const float* b1 = (const float*)&rb[0][1];
#pragma unroll
            for (int j = 0; j < 4; j++) {
                float2 pa; pa.x = a0[j]; pa.y = a1[j];
                *(float2*)&As2[rp0][c40 + j][0] = pa;
                float2 pb; pb.x = b0[j]; pb.y = b1[j];
                *(float2*)&Bs2[rp0][c40 + j][0] = pb;
            }
            const float* a2 = (const float*)&ra[1][0];
            const float* a3 = (const float*)&ra[1][1];
            const float* b2 = (const float*)&rb[1][0];
            const float* b3 = (const float*)&rb[1][1];
#pragma unroll
            for (int j = 0; j < 4; j++) {
                float2 pa; pa.x = a2[j]; pa.y = a3[j];
                *(float2*)&As2[rp1][c41 + j][0] = pa;
                float2 pb; pb.x = b2[j]; pb.y = b3[j];
                *(float2*)&Bs2[rp1][c41 + j][0] = pb;
            }
        }
        __syncthreads();

        // prefetch next stage into registers (overlaps with WMMA burst below)
        const int kn = kk + KC;
        if (kn < FOLD) {
            ra[0][0] = *(const float4*)(Xf + (size_t)(kn + 2 * rp0) * D + m0 + c40);
            ra[0][1] = *(const float4*)(Xf + (size_t)(kn + 2 * rp0 + 1) * D + m0 + c40);
            ra[1][0] = *(const float4*)(Xf + (size_t)(kn + 2 * rp1) * D + m0 + c41);
            ra[1][1] = *(const float4*)(Xf + (size_t)(kn + 2 * rp1 + 1) * D + m0 + c41);
            rb[0][0] = *(const float4*)(Bsrc + (size_t)(kn + 2 * rp0) * ldb + bcol0 + c40);
            rb[0][1] = *(const float4*)(Bsrc + (size_t)(kn + 2 * rp0 + 1) * ldb + bcol0 + c40);
            rb[1][0] = *(const float4*)(Bsrc + (size_t)(kn + 2 * rp1) * ldb + bcol0 + c41);
            rb[1][1] = *(const float4*)(Bsrc + (size_t)(kn + 2 * rp1 + 1) * ldb + bcol0 + c41);
        }

#pragma unroll
        for (int k4 = 0; k4 < KC; k4 += 4) {
            const int kp = (k4 >> 1) + half;   // K-pair index for this lane half
            v2f a[4], b[2];
#pragma unroll
            for (int mi = 0; mi < 4; mi++)
                a[mi] = *(const v2f*)&As2[kp][wm * 64 + mi * 16 + lm][0];
#pragma unroll
            for (int ni = 0; ni < 2; ni++)
                b[ni] = *(const v2f*)&Bs2[kp][wn * 32 + ni * 16 + lm][0];
#pragma unroll
            for (int mi = 0; mi < 4; mi++)
#pragma unroll
                for (int ni = 0; ni < 2; ni++)
                    acc[mi][ni] = wmma4(a[mi], b[ni], acc[mi][ni]);
        }
    }
    float* out = isY ? (xyT + (size_t)fold * D * O) : (xxT + (size_t)fold * D * D);
    const int ldo = isY ? O : D;
#pragma unroll
    for (int mi = 0; mi < 4; mi++)
#pragma unroll
        for (int ni = 0; ni < 2; ni++)
#pragma unroll
            for (int v = 0; v < 8; v++) {
                int row = m0 + wm * 64 + mi * 16 + v + 8 * half;
                int col = bcol0 + wn * 32 + ni * 16 + lm;
                out[(size_t)row * ldo + col] = acc[mi][ni][v];
            }
}

// ---------------------------------------------------------------------------
// Per-fold column sums (and sum of y^2)
// ---------------------------------------------------------------------------
__global__ void sumx_kernel(const float* __restrict__ X, float* __restrict__ sumX_f) {
    int fold = blockIdx.x, d = threadIdx.x;      // 512 threads
    const float* Xf = X + (size_t)fold * FOLD * D;
    float s = 0.f;
    for (int r = 0; r < FOLD; r++) s += Xf[(size_t)r * D + d];
    sumX_f[fold * D + d] = s;
}

__global__ void sumy_kernel(const float* __restrict__ Y, float* __restrict__ sumY_f,
                            float* __restrict__ yy_f) {
    int fold = blockIdx.x, o = threadIdx.x;      // 256 threads
    const float* Yf = Y + (size_t)fold * FOLD * O;
    float s = 0.f, q = 0.f;
    for (int r = 0; r < FOLD; r++) { float v = Yf[(size_t)r * O + o]; s += v; q += v * v; }
    sumY_f[fold * O + o] = s;
    __shared__ float red[256];
    red[o] = q; __syncthreads();
    for (int st = 128; st > 0; st >>= 1) { if (o < st) red[o] += red[o + st]; __syncthreads(); }
    if (o == 0) yy_f[fold] = red[0];
}

__global__ void mu_kernel(const float* __restrict__ sumX_f, const float* __restrict__ sumY_f,
                          float* __restrict__ muX, float* __restrict__ muY) {
    int k = blockIdx.x, t = threadIdx.x;         // 512 threads
    float sx = 0.f;
    for (int f = 0; f < NF; f++) sx += sumX_f[f * D + t];
    muX[k * D + t] = (sx - sumX_f[k * D + t]) / NTR;
    if (t < O) {
        float sy = 0.f;
        for (int f = 0; f < NF; f++) sy += sumY_f[f * O + t];
        muY[k * O + t] = (sy - sumY_f[k * O + t]) / NTR;
    }
}

// ---------------------------------------------------------------------------
// Build batched A = leave-one-fold-out centered XTX + lam*I, and centered XTy
// ---------------------------------------------------------------------------
__global__ void buildA_kernel(const float* __restrict__ xxT_f, const float* __restrict__ muX,
                              float* __restrict__ Abat) {
    int sys = blockIdx.y, l = sys / NF, k = sys % NF;
    int d = blockIdx.x, e = threadIdx.x;         // 512 threads
    float s = 0.f;
    for (int f = 0; f < NF; f++) s += xxT_f[((size_t)f * D + d) * D + e];
    s -= xxT_f[((size_t)k * D + d) * D + e];
    s -= NTR * muX[k * D + d] * muX[k * D + e];
    if (d == e) s += c_lam[l];
    Abat[((size_t)sys * D + d) * D + e] = s;
}

__global__ void buildB_kernel(const float* __restrict__ xyT_f, const float* __restrict__ muX,
                              const float* __restrict__ muY, float* __restrict__ Bc) {
    int k = blockIdx.y, d = blockIdx.x, o = threadIdx.x;   // 256 threads
    float s = 0.f;
    for (int f = 0; f < NF; f++) s += xyT_f[((size_t)f * D + d) * O + o];
    s -= xyT_f[((size_t)k * D + d) * O + o];
    s -= NTR * muX[k * D + d] * muY[k * O + o];
    Bc[((size_t)k * D + d) * O + o] = s;
}

// ---------------------------------------------------------------------------
// Batched in-place Cholesky (lower), 1 workgroup per SPD 512x512 matrix
// ---------------------------------------------------------------------------
__global__ void chol_kernel(float* __restrict__ Abat) {
    float* A = Abat + (size_t)blockIdx.x * D * D;
    int t = threadIdx.x;                          // 256 threads
    __shared__ float sinv;
    for (int j = 0; j < D; j++) {
        __syncthreads();
        if (t == 0) {
            float dg = sqrtf(A[(size_t)j * D + j]);
            A[(size_t)j * D + j] = dg;
            sinv = 1.0f / dg;
        }
        __syncthreads();
        float inv = sinv;
        for (int i = j + 1 + t; i < D; i += 256) A[(size_t)i * D + j] *= inv;
        __syncthreads();
        for (int e = j + 1; e < D; e++) {
            float le = A[(size_t)e * D + j];
            for (int i = e + t; i < D; i += 256)
                A[(size_t)i * D + e] -= A[(size_t)i * D + j] * le;
        }
    }
}

// ---------------------------------------------------------------------------
// Triangular solves L L^T W = B; one thread per RHS column
// ---------------------------------------------------------------------------
__global__ void solve_kernel(const float* __restrict__ Abat, const float* __restrict__ Bc,
                             float* __restrict__ Wbat, int shareB) {
    int sys = blockIdx.x;
    int k = shareB ? (sys % NF) : 0;
    const float* L = Abat + (size_t)sys * D * D;
    const float* B = Bc + (size_t)k * D * O;
    float* W = Wbat + (size_t)sys * D * O;
    int o = threadIdx.x;                          // 256 threads
    for (int j = 0; j < D; j++) {                 // forward: L z = b
        float s = B[(size_t)j * O + o];
        for (int i = 0; i < j; i++) s -= L[(size_t)j * D + i] * W[(size_t)i * O + o];
        W[(size_t)j * O + o] = s / L[(size_t)j * D + j];
    }
    for (int j = D - 1; j >= 0; j--) {            // backward: L^T w = z
        float s = W[(size_t)j * O + o];
        for (int i = j + 1; i < D; i++) s -= L[(size_t)i * D + j] * W[(size_t)i * O + o];
        W[(size_t)j * O + o] = s / L[(size_t)j * D + j];
    }
}

// ---------------------------------------------------------------------------
// b[l,k] = muY[k] - W^T muX[k];  tsx = W^T sumX_f[k]  (validation-side term)
// ---------------------------------------------------------------------------
__global__ void bvec_kernel(const float* __restrict__ Wbat, const float* __restrict__ muX,
                            const float* __restrict__ muY, const float* __restrict__ sumX_f,
                            float* __restrict__ bvec, float* __restrict__ tsx) {
    int sys = blockIdx.x, k = sys % NF, o = threadIdx.x;
    const float* W = Wbat + (size_t)sys * D * O;
    float sb = 0.f, st = 0.f;
    for (int d = 0; d < D; d++) {
        float w = W[(size_t)d * O + o];
        sb += w * muX[k * D + d];
        st += w * sumX_f[k * D + d];
    }
    bvec[sys * O + o] = muY[k * O + o] - sb;
    tsx[sys * O + o] = st;
}

// ---------------------------------------------------------------------------
// q = <W, Gv W> and r = <W, Hv> per system, via WMMA (Gv symmetric -> coalesced A)
// ---------------------------------------------------------------------------
__global__ void lossquad_kernel(const float* __restrict__ xxT_f, const float* __restrict__ xyT_f,
                                const float* __restrict__ Wbat, float* __restrict__ qr) {
    int sys = blockIdx.x, k = sys % NF;
    const float* G = xxT_f + (size_t)k * D * D;
    const float* H = xyT_f + (size_t)k * D * O;
    const float* W = Wbat + (size_t)sys * D * O;
    int t = threadIdx.x, wave = t >> 5, lane = t & 31, half = lane >> 4, lm = lane & 15;
    int n0 = wave * 32;
    float q = 0.f, r = 0.f;
    for (int m0 = 0; m0 < D; m0 += 16) {
        v8f acc0 = {}, acc1 = {};
        for (int kk = 0; kk < D; kk += 4) {
            int k0 = kk + 2 * half, k1 = k0 + 1;
            v2f a, b0, b1;
            a.x = G[(size_t)k0 * D + m0 + lm];          // G[m][k] == G[k][m]
            a.y = G[(size_t)k1 * D + m0 + lm];
            b0.x = W[(size_t)k0 * O + n0 + lm];
            b0.y = W[(size_t)k1 * O + n0 + lm];
            b1.x = W[(size_t)k0 * O + n0 + 16 + lm];
            b1.y = W[(size_t)k1 * O + n0 + 16 + lm];
            acc0 = wmma4(a, b0, acc0);
            acc1 = wmma4(a, b1, acc1);
        }
#pragma unroll
        for (int v = 0; v < 8; v++) {
            int row = m0 + v + 8 * half;
            int c0 = n0 + lm, c1 = n0 + 16 + lm;
            float w0 = W[(size_t)row * O + c0], w1 = W[(size_t)row * O + c1];
            q += acc0[v] * w0 + acc1[v] * w1;
            r += H[(size_t)row * O + c0] * w0 + H[(size_t)row * O + c1] * w1;
        }
    }
    __shared__ float rq[256], rr[256];
    rq[t] = q; rr[t] = r; __syncthreads();
    for (int st = 128; st > 0; st >>= 1) {
        if (t < st) { rq[t] += rq[t + st]; rr[t] += rr[t + st]; }
        __syncthreads();
    }
    if (t == 0) { qr[sys * 2 + 0] = rq[0]; qr[sys * 2 + 1] = rr[0]; }
}

// ---------------------------------------------------------------------------
// losses + argmin (single block)
// ---------------------------------------------------------------------------
__global__ void lossfin_kernel(const float* __restrict__ qr, const float* __restrict__ bvec,
                               const float* __restrict__ tsx, const float* __restrict__ sumY_f,
                               const float* __restrict__ yy_f, float* __restrict__ losses,
                               int* __restrict__ lamIdx, float* __restrict__ out_losses) {
    int t = threadIdx.x;                          // 32 threads
    __shared__ float part[32];
    if (t < NL * NF) {
        int k = t % NF;
        float sbt = 0.f, sbb = 0.f, sbsy = 0.f;
        for (int o = 0; o < O; o++) {
            float b = bvec[t * O + o];
            sbt += b * tsx[t * O + o];
            sbb += b * b;
            sbsy += b * sumY_f[k * O + o];
        }
        part[t] = qr[t * 2] + 2.f * sbt + (float)FOLD * sbb - 2.f * qr[t * 2 + 1]
                  - 2.f * sbsy + yy_f[k];
    }
    __syncthreads();
    if (t == 0) {
        float best = 3.4e38f; int bi = 0;
        for (int l = 0; l < NL; l++) {
            float s = 0.f;
            for (int k = 0; k < NF; k++) s += part[l * NF + k];
            s /= ((float)NF * (float)FOLD * (float)O);
            losses[l] = s; out_losses[l] = s;
            if (s < best) { best = s; bi = l; }
        }
        *lamIdx = bi;
    }
}

// ---------------------------------------------------------------------------
// Final full-data system with CV-optimal lambda
// ---------------------------------------------------------------------------
__global__ void buildAfull_kernel(const float* __restrict__ xxT_f, const float* __restrict__ sumX_f,
                                  const int* __restrict__ lamIdx, float* __restrict__ Afull) {
    int d = blockIdx.x, e = threadIdx.x;          // 512 threads
    float s = 0.f;
    for (int f = 0; f < NF; f++) s += xxT_f[((size_t)f * D + d) * D + e];
    float sxd = 0.f, sxe = 0.f;
    for (int f = 0; f < NF; f++) { sxd += sumX_f[f * D + d]; sxe += sumX_f[f * D + e]; }
    s -= sxd * sxe / NALL;                        // N * mu_d * mu_e
    if (d == e) s += c_lam[*lamIdx];
    Afull[(size_t)d * D + e] = s;
}

__global__ void buildBfull_kernel(const float* __restrict__ xyT_f, const float* __restrict__ sumX_f,
                                  const float* __restrict__ sumY_f, float* __restrict__ Bfull) {
    int d = blockIdx.x, o = threadIdx.x;          // 256 threads
    float s = 0.f;
    for (int f = 0; f < NF; f++) s += xyT_f[((size_t)f * D + d) * O + o];
    float sxd = 0.f, syo = 0.f;
    for (int f = 0; f < NF; f++) { sxd += sumX_f[f * D + d]; syo += sumY_f[f * O + o]; }
    s -= sxd * syo / NALL;
    Bfull[(size_t)d * O + o] = s;
}

__global__ void output_kernel(const float* __restrict__ Wfull, const float* __restrict__ sumX_f,
                              const float* __restrict__ sumY_f, float* __restrict__ out) {
    int o = blockIdx.x, d = threadIdx.x;          // 512 threads
    float w = Wfull[(size_t)d * O + o];
    out[(size_t)o * D + d] = w;                   // w_full.T  [O, D]
    float sxd = 0.f;
    for (int f = 0; f < NF; f++) sxd += sumX_f[f * D + d];
    __shared__ float red[512];
    red[d] = w * (sxd / NALL);
    __syncthreads();
    for (int st = 256; st > 0; st >>= 1) { if (d < st) red[d] += red[d + st]; __syncthreads(); }
    if (d == 0) {
        float syo = 0.f;
        for (int f = 0; f < NF; f++) syo += sumY_f[f * O + o];
        out[(size_t)O * D + o] = syo / NALL - red[0];   // b_full
    }
}

// ---------------------------------------------------------------------------
extern "C" void kernel_launch(void* const* d_in, const int* in_sizes, int n_in,
                              void* d_out, int out_size, void* d_ws, size_t ws_size,
                              hipStream_t stream) {
    (void)in_sizes; (void)n_in; (void)out_size; (void)ws_size;
    const float* X = (const float*)d_in[0];   // [50000, 512]
    const float* Y = (const float*)d_in[1];   // [50000, 256]
    float* out = (float*)d_out;               // [256*512 + 256 + 5]

    float* ws = (float*)d_ws;
    size_t off = 0;
    float* xxT_f  = ws + off; off += (size_t)NF * D * D;       // 1,310,720
    float* xyT_f  = ws + off; off += (size_t)NF * D * O;       //   655,360
    float* sumX_f = ws + off; off += NF * D;
    float* sumY_f = ws + off; off += NF * O;
    float* yy_f   = ws + off; off += 8;
    float* muX    = ws + off; off += NF * D;
    float* muY    = ws + off; off += NF * O;
    float* Abat   = ws + off; off += (size_t)NL * NF * D * D;  // 6,553,600
    float* Bc     = ws + off; off += (size_t)NF * D * O;
    float* Wbat   = ws + off; off += (size_t)NL * NF * D * O;  // 3,276,800
    float* bvec   = ws + off; off += NL * NF * O;
    float* tsx    = ws + off; off += NL * NF * O;
    float* qr     = ws + off; off += 64;
    float* losses = ws + off; off += 8;
    int*   lamIdx = (int*)(ws + off); off += 8;
    float* Afull  = ws + off; off += (size_t)D * D;
    float* Bfull  = ws + off; off += (size_t)D * O;
    float* Wfull  = ws + off; off += (size_t)D * O;

    // 1. per-fold Gram matrices via fp32 WMMA
    stats_kernel<<<dim3(6, 4, NF), 256, 0, stream>>>(X, Y, xxT_f, xyT_f);
    // 2. per-fold sums / means
    sumx_kernel<<<NF, D, 0, stream>>>(X, sumX_f);
    sumy_kernel<<<NF, O, 0, stream>>>(Y, sumY_f, yy_f);
    mu_kernel<<<NF, D, 0, stream>>>(sumX_f, sumY_f, muX, muY);
    // 3. batched ridge systems
    buildA_kernel<<<dim3(D, NL * NF), D, 0, stream>>>(xxT_f, muX, Abat);
    buildB_kernel<<<dim3(D, NF), O, 0, stream>>>(xyT_f, muX, muY, Bc);
    chol_kernel<<<NL * NF, 256, 0, stream>>>(Abat);
    solve_kernel<<<NL * NF, 256, 0, stream>>>(Abat, Bc, Wbat, 1);
    // 4. CV losses from sufficient statistics (WMMA for <W, Gv W>)
    bvec_kernel<<<NL * NF, O, 0, stream>>>(Wbat, muX, muY, sumX_f, bvec, tsx);
    lossquad_kernel<<<NL * NF, 256, 0, stream>>>(xxT_f, xyT_f, Wbat, qr);
    lossfin_kernel<<<1, 32, 0, stream>>>(qr, bvec, tsx, sumY_f, yy_f, losses,
                                         lamIdx, out + (size_t)O * D + O);
    // 5. final full-data fit with argmin lambda
    buildAfull_kernel<<<D, D, 0, stream>>>(xxT_f, sumX_f, lamIdx, Afull);
    buildBfull_kernel<<<D, O, 0, stream>>>(xyT_f, sumX_f, sumY_f, Bfull);
    chol_kernel<<<1, 256, 0, stream>>>(Afull);
    solve_kernel<<<1, 256, 0, stream>>>(Afull, Bfull, Wfull, 0);
    output_kernel<<<O, D, 0, stream>>>(Wfull, sumX_f, sumY_f, out);
}